// TapasBaselineModel_57080115363954
// MI455X (gfx1250) — compile-verified
//
#include <hip/hip_runtime.h>

// ---------------------------------------------------------------------------
// TAPAS baseline heads on MI455X (gfx1250, wave32).
//
// Roofline: naive edge head = 79872x1536x256 GEMM (63 GFLOP). Since the edge
// feature is concat(cell[e0], cell[e1]), it factors into
//   Q = cell(2048x768) @ [We1_top | We1_bot](768x512)   (1.6 GFLOP, f32 WMMA)
//   edge_logit = relu(Q0[e0] + Q1[e1] + bias') . We2    (82 MFLOP, L2-bound)
// Q is 4 MB -> fully L2-resident (192 MB); the ~163 MB of per-edge Q traffic
// never touches HBM. BN (eval) folds into column scales + one 256-wide bias.
// GEMMs use V_WMMA_F32_16X16X4_F32; the shared 16x768 A-tile is staged in LDS
// via GLOBAL_LOAD_ASYNC_TO_LDS_B128 (ASYNCcnt) and read via ds_load.
// ---------------------------------------------------------------------------

#define BB   8
#define SS   512
#define DD   768
#define RR   32
#define CCC  8
#define RCN  256        // R*C cells per table
#define EE   9984       // edges per table
#define HH   256        // hidden width
#define NRC  320        // B*(R+C)
#define NEDGE (BB*EE)   // 79872
#define BN_EPS 1e-5f
#define ASTR 772        // padded LDS row stride (772 % 64 = 4 -> conflict-free)

typedef __attribute__((ext_vector_type(2))) float v2f;
typedef __attribute__((ext_vector_type(8))) float v8f;
typedef __attribute__((ext_vector_type(4))) int   v4i_t;

#define AS1 __attribute__((address_space(1)))
#define AS3 __attribute__((address_space(3)))

#if defined(__HIP_DEVICE_COMPILE__) && __has_builtin(__builtin_amdgcn_global_load_async_to_lds_b128)
#define HAVE_ASYNC_LDS 1
#else
#define HAVE_ASYNC_LDS 0
#endif

// Stage a 16 x 768 f32 A-tile into LDS, coalesced, 16B per lane per step.
// On gfx1250 this uses GLOBAL_LOAD_ASYNC_TO_LDS_B128 (tracked by ASYNCcnt, no
// VGPR round-trip); each wave drains its own ASYNCcnt, then the barrier
// publishes the tile to all 4 waves.
__device__ __forceinline__ void stage_a_tile(const float* __restrict__ Atile,
                                             float* __restrict__ As) {
    const int tid = threadIdx.x;                    // 128 threads
    for (int i = tid; i < 16 * 192; i += 128) {     // 192 float4 chunks per row
        int row = i / 192;
        int c4  = i - row * 192;
        const float* gsrc = Atile + (size_t)row * DD + c4 * 4;
        float*       ldst = As + row * ASTR + c4 * 4;
#if HAVE_ASYNC_LDS
        // addrspace(1)/(3) pointers synthesized via integer casts: flat-addr
        // low 32 bits == LDS byte offset (ISA 10.2 aperture mapping).
        __builtin_amdgcn_global_load_async_to_lds_b128(
            (AS1 v4i_t*)(size_t)gsrc,
            (AS3 v4i_t*)(unsigned int)(size_t)ldst, 0, 0);
#else
        *(float4*)ldst = *(const float4*)gsrc;
#endif
    }
#if HAVE_ASYNC_LDS
#if __has_builtin(__builtin_amdgcn_s_wait_asynccnt)
    __builtin_amdgcn_s_wait_asynccnt(0);
#else
    asm volatile("s_wait_asynccnt 0x0" ::: "memory");
#endif
#endif
    __syncthreads();
}

// ---------------------------------------------------------------------------
// Fold BatchNorm(eval) into per-channel scale a[] and bias c[]:
//   bn(x + b) = (x + b - m) * g * rsqrt(v+eps) + bt  =  x*a + ((b-m)*a + bt)
// ---------------------------------------------------------------------------
__global__ void k_prep(const float* __restrict__ b1,  const float* __restrict__ g1,
                       const float* __restrict__ bt1, const float* __restrict__ m1,
                       const float* __restrict__ v1,
                       const float* __restrict__ be1, const float* __restrict__ ge1,
                       const float* __restrict__ bte1,const float* __restrict__ me1,
                       const float* __restrict__ ve1,
                       float* __restrict__ a1, float* __restrict__ c1,
                       float* __restrict__ ae, float* __restrict__ ce) {
    int j = threadIdx.x;                 // 256 threads
    float a = g1[j] * rsqrtf(v1[j] + BN_EPS);
    a1[j] = a;
    c1[j] = (b1[j] - m1[j]) * a + bt1[j];
    float ax = ge1[j] * rsqrtf(ve1[j] + BN_EPS);
    ae[j] = ax;
    ce[j] = (be1[j] - me1[j]) * ax + bte1[j];
}

// ---------------------------------------------------------------------------
// Gather first-token embedding of every cell: cell[b*256+i, :] = h[b, idx, :]
// ---------------------------------------------------------------------------
__global__ void k_gather_cell(const float* __restrict__ h, const int* __restrict__ idx,
                              float* __restrict__ cell) {
    int blk = blockIdx.x;                // 0..2047
    int b = blk >> 8;
    int i = blk & 255;
    int s = idx[b * RCN + i];
    const float* src = h + ((size_t)b * SS + (size_t)s) * DD;
    float* dst = cell + (size_t)blk * DD;
    for (int d = threadIdx.x; d < DD; d += blockDim.x) dst[d] = src[d];
}

// ---------------------------------------------------------------------------
// rc rows: per table, 32 row-means (over 8 cols) then 8 col-means (over the
// 31 valid rows). rc is [320, 768].
// ---------------------------------------------------------------------------
__global__ void k_build_rc(const float* __restrict__ cell, float* __restrict__ rc) {
    int g = blockIdx.x;                  // 0..319
    int b = g / (RR + CCC);
    int j = g - b * (RR + CCC);
    const float* base = cell + (size_t)b * RCN * DD;
    float* dst = rc + (size_t)g * DD;
    for (int d = threadIdx.x; d < DD; d += blockDim.x) {
        float acc = 0.f;
        if (j < RR) {
            const float* p = base + (size_t)(j * CCC) * DD + d;
            #pragma unroll
            for (int c = 0; c < CCC; ++c) acc += p[(size_t)c * DD];
            acc *= (1.0f / CCC);
        } else {
            int jc = j - RR;
            const float* p = base + (size_t)jc * DD + d;
            for (int r = 0; r < RR - 1; ++r) acc += p[(size_t)(r * CCC) * DD];
            acc *= (1.0f / (RR - 1));
        }
        dst[d] = acc;
    }
}

// ---------------------------------------------------------------------------
// WMMA f32 16x16x4 GEMM: Q[2048,512] = cell[2048,768] @ [We1_top|We1_bot],
// columns scaled by folded BN scale ae[n&255]. 4 waves/block share one LDS
// A-tile; one 16x16 output tile per wave, 192 v_wmma per tile.
//
// Fragment layout (ISA 7.12.2, 32-bit A 16x4): lane<16 -> row M=lane, K=0,1;
// lane>=16 -> row M=lane-16, K=2,3. B mirrors with N=lane&15. C/D: VGPR r
// holds M=r (lanes 0-15) and M=r+8 (lanes 16-31), N=lane&15.
// ---------------------------------------------------------------------------
__global__ void k_gemm_q(const float* __restrict__ cell, const float* __restrict__ We1,
                         const float* __restrict__ ae, float* __restrict__ Q) {
    __shared__ float As[16 * ASTR];
    const int lane = threadIdx.x & 31;
    const int wave = threadIdx.x >> 5;
    const int mt = blockIdx.x;                    // 0..127
    const int nt = blockIdx.y * 4 + wave;         // 0..31
    const int lm = lane & 15;
    const int khi = (lane >> 4) << 1;             // 0 or 2
    const int ncol = nt * 16 + lm;                // 0..511
    const int nb = ncol & 255;
    const float* Bp = We1 + (nt >= 16 ? (size_t)DD * HH : (size_t)0);

    stage_a_tile(cell + (size_t)(mt * 16) * DD, As);
    const float* Arow = As + lm * ASTR;

    v8f acc = {0.f,0.f,0.f,0.f,0.f,0.f,0.f,0.f};
    for (int kk = 0; kk < DD; kk += 4) {
        v2f a, b;
        a.x = Arow[kk + khi];
        a.y = Arow[kk + khi + 1];
        b.x = Bp[(size_t)(kk + khi)     * HH + nb];
        b.y = Bp[(size_t)(kk + khi + 1) * HH + nb];
        acc = __builtin_amdgcn_wmma_f32_16x16x4_f32(
                  false, a, false, b, (short)0, acc, false, false);
    }
    const float scale = ae[nb];
    const int mbase = mt * 16 + ((lane >> 4) << 3);
    #pragma unroll
    for (int r = 0; r < 8; ++r)
        Q[(size_t)(mbase + r) * 512 + ncol] = acc[r] * scale;
}

// ---------------------------------------------------------------------------
// WMMA GEMM + fused BN affine + ReLU: Hrc[320,256] = relu(a1*(rc@W1) + c1)
// ---------------------------------------------------------------------------
__global__ void k_gemm_rc(const float* __restrict__ rc, const float* __restrict__ W1,
                          const float* __restrict__ a1, const float* __restrict__ c1,
                          float* __restrict__ Hrc) {
    __shared__ float As[16 * ASTR];
    const int lane = threadIdx.x & 31;
    const int wave = threadIdx.x >> 5;
    const int mt = blockIdx.x;                    // 0..19
    const int nt = blockIdx.y * 4 + wave;         // 0..15
    const int lm = lane & 15;
    const int khi = (lane >> 4) << 1;
    const int ncol = nt * 16 + lm;                // 0..255

    stage_a_tile(rc + (size_t)(mt * 16) * DD, As);
    const float* Arow = As + lm * ASTR;

    v8f acc = {0.f,0.f,0.f,0.f,0.f,0.f,0.f,0.f};
    for (int kk = 0; kk < DD; kk += 4) {
        v2f a, b;
        a.x = Arow[kk + khi];
        a.y = Arow[kk + khi + 1];
        b.x = W1[(size_t)(kk + khi)     * HH + ncol];
        b.y = W1[(size_t)(kk + khi + 1) * HH + ncol];
        acc = __builtin_amdgcn_wmma_f32_16x16x4_f32(
                  false, a, false, b, (short)0, acc, false, false);
    }
    const float sa = a1[ncol], sc = c1[ncol];
    const int mbase = mt * 16 + ((lane >> 4) << 3);
    #pragma unroll
    for (int r = 0; r < 8; ++r)
        Hrc[(size_t)(mbase + r) * HH + ncol] = fmaxf(acc[r] * sa + sc, 0.f);
}

// ---------------------------------------------------------------------------
// Edge head: one wave per edge (8 waves/block). ce/We2 staged in LDS and
// reused by all 8 waves. Each lane handles 8 channels:
//   s = sum_j relu(Q[e0,j] + Q[e1,256+j] + ce[j]) * We2[j]
// then shfl-xor wave reduction; mask -> -inf.
// ---------------------------------------------------------------------------
__global__ void k_edge(const float* __restrict__ Q, const int* __restrict__ edges,
                       const unsigned char* __restrict__ emask,
                       const float* __restrict__ ce, const float* __restrict__ We2,
                       const float* __restrict__ be2, float* __restrict__ out) {
    __shared__ float sce[HH];
    __shared__ float swe[HH];
    sce[threadIdx.x] = ce[threadIdx.x];            // blockDim == 256 == HH
    swe[threadIdx.x] = We2[threadIdx.x];
    __syncthreads();

    const int wave = threadIdx.x >> 5;
    const int lane = threadIdx.x & 31;
    const int t = blockIdx.x * (blockDim.x >> 5) + wave;   // 0..79871 exactly
    const int b = t / EE;
    const int e = t - b * EE;
    const int e0 = edges[2 * e];
    const int e1 = edges[2 * e + 1];
    const int j0 = lane << 3;                               // 8 channels per lane
    const float* q0 = Q + (size_t)(b * RCN + e0) * 512 + j0;
    const float* q1 = Q + (size_t)(b * RCN + e1) * 512 + HH + j0;
    float s = 0.f;
    #pragma unroll
    for (int u = 0; u < 8; ++u) {
        float v = fmaxf(q0[u] + q1[u] + sce[j0 + u], 0.f);
        s = fmaf(v, swe[j0 + u], s);
    }
    #pragma unroll
    for (int off = 16; off > 0; off >>= 1)
        s += __shfl_xor(s, off, 32);
    if (lane == 0) {
        float r = s + be2[0];
        if (emask[t]) r = __int_as_float(0xff800000);       // -inf
        out[t] = r;
    }
}

// ---------------------------------------------------------------------------
// rc output: logits[row, o] = Hrc[row,:] . W2[:,o] + b2[o]; padding rows get
// the fixed override [1,0,0,0].
// ---------------------------------------------------------------------------
__global__ void k_rc_out(const float* __restrict__ Hrc, const float* __restrict__ W2,
                         const float* __restrict__ b2, const unsigned char* __restrict__ rmask,
                         float* __restrict__ out) {
    int tid = blockIdx.x * blockDim.x + threadIdx.x;        // 0..1279
    if (tid >= NRC * 4) return;
    int row = tid >> 2;
    int o = tid & 3;
    float r;
    if (rmask[row]) {
        r = (o == 0) ? 1.0f : 0.0f;
    } else {
        float s = b2[o];
        const float* hp = Hrc + (size_t)row * HH;
        #pragma unroll 8
        for (int k = 0; k < HH; ++k)
            s = fmaf(hp[k], W2[k * 4 + o], s);
        r = s;
    }
    out[tid] = r;
}

// ---------------------------------------------------------------------------
extern "C" void kernel_launch(void* const* d_in, const int* in_sizes, int n_in,
                              void* d_out, int out_size, void* d_ws, size_t ws_size,
                              hipStream_t stream) {
    (void)in_sizes; (void)n_in; (void)out_size; (void)ws_size;
    const float* h    = (const float*)d_in[0];
    const int*   idx  = (const int*)d_in[1];
    const int*   edges= (const int*)d_in[2];
    const unsigned char* rmask = (const unsigned char*)d_in[3];
    const unsigned char* emask = (const unsigned char*)d_in[4];
    const float* W1   = (const float*)d_in[5];
    const float* b1   = (const float*)d_in[6];
    const float* g1   = (const float*)d_in[7];
    const float* bt1  = (const float*)d_in[8];
    const float* m1   = (const float*)d_in[9];
    const float* v1   = (const float*)d_in[10];
    const float* W2   = (const float*)d_in[11];
    const float* b2   = (const float*)d_in[12];
    const float* We1  = (const float*)d_in[13];
    const float* be1  = (const float*)d_in[14];
    const float* ge1  = (const float*)d_in[15];
    const float* bte1 = (const float*)d_in[16];
    const float* me1  = (const float*)d_in[17];
    const float* ve1  = (const float*)d_in[18];
    const float* We2  = (const float*)d_in[19];
    const float* be2  = (const float*)d_in[20];

    // Workspace layout (floats); total ~2.95M floats = 11.8 MB
    float* ws   = (float*)d_ws;
    float* cell = ws;                            // 2048*768
    float* Q    = cell + 2048 * 768;             // 2048*512
    float* rc   = Q    + 2048 * 512;             // 320*768
    float* Hrc  = rc   + NRC  * 768;             // 320*256
    float* a1   = Hrc  + NRC  * HH;              // 256
    float* c1   = a1 + HH;                       // 256
    float* ae   = c1 + HH;                       // 256
    float* ce   = ae + HH;                       // 256

    float* out_rc = (float*)d_out;               // [320, 4]
    float* out_e  = out_rc + NRC * 4;            // [79872]

    k_prep<<<1, 256, 0, stream>>>(b1, g1, bt1, m1, v1,
                                  be1, ge1, bte1, me1, ve1,
                                  a1, c1, ae, ce);
    k_gather_cell<<<2048, 256, 0, stream>>>(h, idx, cell);
    k_build_rc<<<NRC, 256, 0, stream>>>(cell, rc);
    k_gemm_q <<<dim3(128, 8), 128, 0, stream>>>(cell, We1, ae, Q);
    k_gemm_rc<<<dim3(20, 4),  128, 0, stream>>>(rc, W1, a1, c1, Hrc);
    k_edge   <<<NEDGE / 8, 256, 0, stream>>>(Q, edges, emask, ce, We2, be2, out_e);
    k_rc_out <<<5, 256, 0, stream>>>(Hrc, W2, b2, rmask, out_rc);
}